// Primitive_38749194945092
// MI455X (gfx1250) — compile-verified
//
#include <hip/hip_runtime.h>

// Apply a real 2x2 gate P to qubit axis 11 of a (2,)^24 x (4,) f32 statevector.
// Pairs: (i, i + PAIR_STRIDE) within each 2*PAIR_STRIDE super-block.
// Implemented as D = (I8 (x) P) * S with V_WMMA_F32_16X16X4_F32 on 16x16 tiles.

typedef __attribute__((ext_vector_type(2))) float v2f;
typedef __attribute__((ext_vector_type(8))) float v8f;

// stride of target axis 11: 2^(24-1-11) * 4(batch) = 16384 elements
#define PAIR_STRIDE 16384u
#define TILE_COLS   128u          // contiguous floats per side per wave tile
#define NUM_TILES   (1u << 18)    // 2^11 super-blocks * 128 tiles each
                                  // total: 2^18 * 2 * 128 = 2^26 elements

__global__ __launch_bounds__(256) void pauli_apply_wmma(
    const float* __restrict__ state,
    const float* __restrict__ pauli,
    float* __restrict__ out)
{
    const unsigned tid  = blockIdx.x * blockDim.x + threadIdx.x;
    const unsigned wid  = tid >> 5;            // wave id == tile id
    const unsigned lane = threadIdx.x & 31u;
    const unsigned half = lane >> 4;           // 0: lanes 0-15, 1: lanes 16-31
    const unsigned l15  = lane & 15u;          // A row m / B,C,D column n

    // tile -> memory: one 128-float contiguous block on each pair side
    const unsigned upper = wid >> 7;
    const unsigned blk   = wid & 127u;
    const unsigned base0 = upper * (2u * PAIR_STRIDE) + blk * TILE_COLS;
    const unsigned base1 = base0 + PAIR_STRIDE;

    const float p00 = pauli[0], p01 = pauli[1];
    const float p10 = pauli[2], p11 = pauli[3];

    // A fragments: G16 = I8 (x) P sliced into four 16x4 K-chunks.
    // A 16x4 f32 layout: lane m = lane&15 (both halves), VGPR w holds K = w + 2*half.
    v2f a[4];
    {
        const unsigned m = l15;
#pragma unroll
        for (int k = 0; k < 4; ++k) {
#pragma unroll
            for (int w = 0; w < 2; ++w) {
                const unsigned c = 4u * (unsigned)k + (unsigned)w + 2u * half;
                float val = 0.0f;
                if ((c >> 1) == (m >> 1)) {            // inside the 2x2 block
                    const float r0 = (c & 1u) ? p01 : p00;
                    const float r1 = (c & 1u) ? p11 : p10;
                    val = (m & 1u) ? r1 : r0;
                }
                a[k][w] = val;
            }
        }
    }

    // B fragments: tile row m = 4k + w + 2*half, col n = l15.
    // Row m lives at  base_{m&1} + (m>>1)*16 + n  -> 16-lane-contiguous loads.
    v2f b[4];
#pragma unroll
    for (int k = 0; k < 4; ++k) {
#pragma unroll
        for (int w = 0; w < 2; ++w) {
            const unsigned m    = 4u * (unsigned)k + (unsigned)w + 2u * half;
            const unsigned base = (m & 1u) ? base1 : base0;
            const unsigned addr = base + (m >> 1) * 16u + l15;
            b[k][w] = __builtin_nontemporal_load(state + addr);  // streaming: no reuse
        }
    }

    // D = sum_k A_k * B_k  (four v_wmma_f32_16x16x4_f32, accumulate in C)
    v8f acc = {};
#pragma unroll
    for (int k = 0; k < 4; ++k) {
        acc = __builtin_amdgcn_wmma_f32_16x16x4_f32(
            /*neg_a=*/false, a[k], /*neg_b=*/false, b[k],
            /*c_mod=*/(short)0, acc, /*reuse_a=*/false, /*reuse_b=*/false);
    }

    // D 16x16 f32 layout: VGPR v, row M = v + 8*half, col n = l15.
#pragma unroll
    for (int v = 0; v < 8; ++v) {
        const unsigned m    = (unsigned)v + 8u * half;
        const unsigned base = (m & 1u) ? base1 : base0;
        const unsigned addr = base + (m >> 1) * 16u + l15;
        __builtin_nontemporal_store(acc[v], out + addr);  // streaming store (NT)
    }
}

extern "C" void kernel_launch(void* const* d_in, const int* in_sizes, int n_in,
                              void* d_out, int out_size, void* d_ws, size_t ws_size,
                              hipStream_t stream) {
    (void)in_sizes; (void)n_in; (void)out_size; (void)d_ws; (void)ws_size;
    const float* state = (const float*)d_in[0];
    const float* pauli = (const float*)d_in[1];
    float*       out   = (float*)d_out;

    const unsigned threads = 256;
    const unsigned blocks  = (NUM_TILES * 32u) / threads;  // 32768 blocks, 2^23 threads
    hipLaunchKernelGGL(pauli_apply_wmma, dim3(blocks), dim3(threads), 0, stream,
                       state, pauli, out);
}